// Gated_MaxAvgPooling_88802743812918
// MI455X (gfx1250) — compile-verified
//
#include <hip/hip_runtime.h>

// Native ext-vector type so __builtin_nontemporal_load/store lower cleanly
// to global_load_b128 / global_store_b128 with th:TH_*_NT on gfx1250.
typedef float v4f __attribute__((ext_vector_type(4)));

// x: [B=16, H=256, W=256, C=64] fp32 NHWC, mask: [2,2] fp32
// out: [B, H/2, W/2, C] fp32
//
// One thread = one float4 (4 channels) of one output pixel.
// t = [b:4][h:7][w:7][c4:4] (c4 fastest) is exactly the output's
// float4-flattened index -> out4[t].
//
// Input float4 index of x[b, 2h, 2w, 4*c4]:
//   base4 = b*2^20 + h*2^13 + w*2^5 + c4
// i.e. each bit-field of t shifted left in place:
//   base4 = (t & 0xF) | ((t & 0x7F0) << 1) | ((t & ~0x7FF) << 2)
// All offsets fit in 32 bits (input is 256 MiB).
__global__ __launch_bounds__(256) void gated_maxavg_pool_kernel(
    const float* __restrict__ x,
    const float* __restrict__ mask,
    float* __restrict__ out,
    unsigned int n4)
{
    // Uniform address -> scalar loads (s_load), broadcast in SGPRs.
    const float m00 = mask[0];
    const float m01 = mask[1];
    const float m10 = mask[2];
    const float m11 = mask[3];

    const unsigned int t = blockIdx.x * 256u + threadIdx.x;
    if (t >= n4) return;

    // 32-bit field remap: t -> input float4 index.
    const unsigned int base4 = (t & 0xFu)
                             | ((t & 0x7F0u) << 1)
                             | ((t & ~0x7FFu) << 2);

    const v4f* __restrict__ x4 = (const v4f*)x;

    // 2x2 window, 4 channels per lane; row stride = 4096 float4s (64 KiB).
    // Constant offsets fold into the global_load ioffset field.
    // Read-once stream -> NT loads keep the 192MB L2 clean.
    const v4f v00 = __builtin_nontemporal_load(x4 + base4);          // (2h,   2w)
    const v4f v01 = __builtin_nontemporal_load(x4 + base4 + 16);     // (2h,   2w+1)
    const v4f v10 = __builtin_nontemporal_load(x4 + base4 + 4096);   // (2h+1, 2w)
    const v4f v11 = __builtin_nontemporal_load(x4 + base4 + 4112);   // (2h+1, 2w+1)

    v4f res;
    #pragma unroll
    for (int j = 0; j < 4; ++j) {
        const float a  = v00[j];
        const float bb = v01[j];
        const float c  = v10[j];
        const float d  = v11[j];

        const float mx = fmaxf(fmaxf(a, bb), fmaxf(c, d));
        const float av = (a + bb + c + d) * 0.25f;

        // shared 2x2 mask dot (fma chain)
        float xs = a * m00;
        xs = fmaf(bb, m01, xs);
        xs = fmaf(c,  m10, xs);
        xs = fmaf(d,  m11, xs);

        // sigmoid gate: z = 1/(1+e^-xs); v_exp_f32 + v_rcp_f32 (1 ulp each)
        const float e = __expf(-xs);
        const float z = __builtin_amdgcn_rcpf(1.0f + e);

        // z*mx + (1-z)*av == av + z*(mx-av)
        res[j] = fmaf(z, mx - av, av);
    }

    // Write-once stream -> NT store.
    __builtin_nontemporal_store(res, (v4f*)out + t);
}

extern "C" void kernel_launch(void* const* d_in, const int* in_sizes, int n_in,
                              void* d_out, int out_size, void* d_ws, size_t ws_size,
                              hipStream_t stream) {
    const float* x    = (const float*)d_in[0];   // [16,256,256,64] fp32
    const float* mask = (const float*)d_in[1];   // [2,2] fp32
    float* out = (float*)d_out;                  // [16,128,128,64] fp32

    const unsigned int n4 = (unsigned int)(out_size / 4); // 4,194,304 float4s
    const int block = 256;                                // 8 wave32 per block
    const int grid = (int)((n4 + block - 1) / block);     // 16384 blocks
    gated_maxavg_pool_kernel<<<grid, block, 0, stream>>>(x, mask, out, n4);
}